// SpiralConv_36807869727135
// MI455X (gfx1250) — compile-verified
//
#include <hip/hip_runtime.h>
#include <cmath>

#define D_DIM 512
#define B_DIM 4
#define L_DIM 4096
#define CH 16                  // chunk length (WMMA M/K tile)
#define NC (L_DIM / CH)        // 256 chunks per sequence
#define DT 16                  // d-channels per block in WMMA kernel
#define LT 256                 // l per block = 16 chunks = 16 WMMA columns

typedef __attribute__((ext_vector_type(2))) float v2f;
typedef __attribute__((ext_vector_type(8))) float v8f;

// ---- workspace layout (float offsets) ----
#define WS_PWRE 0                                   // ph^i re, [17][D]
#define WS_PWIM (WS_PWRE + (CH + 1) * D_DIM)        // ph^i im, [17][D]
#define WS_TRE  (WS_PWIM + (CH + 1) * D_DIM)        // T re, [D][16][16]
#define WS_TIM  (WS_TRE + D_DIM * CH * CH)          // T im, [D][16][16]
#define WS_E    (WS_TIM + D_DIM * CH * CH)          // carries, interleaved (re,im), B*NC*D pairs

// ---------------------------------------------------------------------------
// K1: per-channel setup: ph = phazor/|phazor| * exp(-|phazor|); powers ph^0..16;
//     lower-triangular chunk matrices T_d[i][j] = init_d * ph_d^{i-j}.
// ---------------------------------------------------------------------------
__global__ void spiral_setup(const float* __restrict__ pr, const float* __restrict__ pi,
                             const float* __restrict__ ir, const float* __restrict__ ii,
                             float* __restrict__ ws) {
  int d = blockIdx.x * blockDim.x + threadIdx.x;
  if (d >= D_DIM) return;
  float a = pr[d], b = pi[d];
  float mag = sqrtf(a * a + b * b);
  float s = expf(-mag) / mag;
  float phr = a * s, phi = b * s;

  // powers ph^0 .. ph^16, stored [i][d] for coalesced reads later
  float cr = 1.f, ci = 0.f;
  for (int i = 0; i <= CH; ++i) {
    ws[WS_PWRE + i * D_DIM + d] = cr;
    ws[WS_PWIM + i * D_DIM + d] = ci;
    float nr = cr * phr - ci * phi;
    float ni = cr * phi + ci * phr;
    cr = nr; ci = ni;
  }

  float vr = ir[d], vi = ii[d];
  float* Tre = ws + WS_TRE + d * (CH * CH);
  float* Tim = ws + WS_TIM + d * (CH * CH);
  for (int i = 0; i < CH; ++i) {
    float tr = vr, ti = vi;                  // init * ph^0 at j == i
    for (int j = i; j >= 0; --j) {
      Tre[i * CH + j] = tr;
      Tim[i * CH + j] = ti;
      float nr = tr * phr - ti * phi;
      float ni = tr * phi + ti * phr;
      tr = nr; ti = ni;
    }
    for (int j = i + 1; j < CH; ++j) { Tre[i * CH + j] = 0.f; Tim[i * CH + j] = 0.f; }
  }
}

// ---------------------------------------------------------------------------
// K2: local chunk scan as f32 WMMA GEMM.  Y[i,n] = sum_j T_d[i][j] * X_d[j,n]
//     (n = chunk column).  A = T_d (16x16 via 4 K-slices), B = x slab from LDS.
//     Writes Y (complex, interleaved) straight into d_out's complex region.
// ---------------------------------------------------------------------------
__global__ void __launch_bounds__(256)
spiral_chunk_wmma(const float* __restrict__ x, const float* __restrict__ ws,
                  float* __restrict__ outc) {
  // x slab: [DT d][16 cols * 17 padded j]  (x17 pad -> conflict-free WMMA B reads)
  __shared__ float xs[DT * (CH * 17)];
  // y slab: [DT d][256 l * 2 comps], pitch 513 to de-conflict the store pass
  __shared__ float ys[DT * (2 * LT + 1)];
  const int YP = 2 * LT + 1;

  int bt = blockIdx.x;
  int db = bt & 31;            // D/DT = 32
  int lb = (bt >> 5) & 15;     // L/LT = 16
  int b  = bt >> 9;            // B = 4
  int d0 = db * DT;
  int l0 = lb * LT;

  // ---- stage x[b, l0:l0+256, d0:d0+16] into LDS, coalesced over d ----
  const size_t xbase = ((size_t)b * L_DIM + l0) * D_DIM + d0;
  int t = threadIdx.x;
  int dcol  = t & (DT - 1);
  int lrow0 = t >> 4;          // 0..15
  for (int it = 0; it < LT / 16; ++it) {
    int lrow = lrow0 + it * 16;
    float v = x[xbase + (size_t)lrow * D_DIM + dcol];
    xs[dcol * (CH * 17) + (lrow >> 4) * 17 + (lrow & (CH - 1))] = v;
  }
  __syncthreads();

  int lane = t & 31;
  int wave = t >> 5;
  int n  = lane & 15;          // WMMA column (chunk within slab)
  int hi = lane >> 4;          // half-wave selector
  int M  = lane & 15;          // A-matrix row

  const float* Tre = ws + WS_TRE;
  const float* Tim = ws + WS_TIM;

  for (int dd = 0; dd < 2; ++dd) {
    int dl = wave * 2 + dd;                      // this wave owns 2 channels
    int d  = d0 + dl;
    const float2* arow_re = (const float2*)(Tre + d * (CH * CH) + M * CH);
    const float2* arow_im = (const float2*)(Tim + d * (CH * CH) + M * CH);
    const float* xrow = xs + dl * (CH * 17) + n * 17;

    v8f accRe = {}; v8f accIm = {};
#pragma unroll
    for (int k = 0; k < 4; ++k) {
      int j0 = 4 * k + 2 * hi;                   // K-slice cols per ISA A layout
      float2 ar = arow_re[2 * k + hi];
      float2 ai = arow_im[2 * k + hi];
      v2f av_re = { ar.x, ar.y };
      v2f av_im = { ai.x, ai.y };
      v2f bv    = { xrow[j0], xrow[j0 + 1] };    // B mirrors A's K striping
      accRe = __builtin_amdgcn_wmma_f32_16x16x4_f32(
          false, av_re, false, bv, (short)0, accRe, false, false);
      accIm = __builtin_amdgcn_wmma_f32_16x16x4_f32(
          false, av_im, false, bv, (short)0, accIm, false, false);
    }
    // C/D layout: VGPR r -> (i = r + 8*hi, col n); stage to LDS for coalesced store
#pragma unroll
    for (int r = 0; r < 8; ++r) {
      int i = r + 8 * hi;
      int l = n * CH + i;
      ys[dl * YP + 2 * l + 0] = accRe[r];
      ys[dl * YP + 2 * l + 1] = accIm[r];
    }
  }
  __syncthreads();

  // ---- coalesced complex store: consecutive lanes -> consecutive d ----
  for (int it = 0; it < 16; ++it) {
    int lrow = lrow0 + it * 16;
    float re = ys[dcol * YP + 2 * lrow + 0];
    float im = ys[dcol * YP + 2 * lrow + 1];
    size_t gi = ((size_t)b * L_DIM + l0 + lrow) * D_DIM + d0 + dcol;
    *(float2*)(outc + 2 * gi) = make_float2(re, im);
  }
}

// ---------------------------------------------------------------------------
// K3: inter-chunk carry scan per (b,d): E[0]=hidden, E[c]=ph^16*E[c-1]+Y15[c-1]
// ---------------------------------------------------------------------------
__global__ void spiral_carry(const float* __restrict__ hr, const float* __restrict__ him,
                             const float* __restrict__ ws, const float* __restrict__ outc,
                             float* __restrict__ E) {
  int g = blockIdx.x * blockDim.x + threadIdx.x;
  if (g >= B_DIM * D_DIM) return;
  int b = g / D_DIM, d = g % D_DIM;
  float p16r = ws[WS_PWRE + CH * D_DIM + d];
  float p16i = ws[WS_PWIM + CH * D_DIM + d];
  float er = hr[g], ei = him[g];
  for (int c = 0; c < NC; ++c) {
    size_t eidx = (size_t)(b * NC + c) * D_DIM + d;
    E[2 * eidx + 0] = er;
    E[2 * eidx + 1] = ei;
    size_t yi = ((size_t)b * L_DIM + c * CH + (CH - 1)) * D_DIM + d;
    float yr = outc[2 * yi], ym = outc[2 * yi + 1];
    float nr = p16r * er - p16i * ei + yr;
    float ni = p16r * ei + p16i * er + ym;
    er = nr; ei = ni;
  }
}

// ---------------------------------------------------------------------------
// K4: out = Y + E * ph^{i+1}; emit real array + corrected complex (interleaved)
// ---------------------------------------------------------------------------
__global__ void __launch_bounds__(256)
spiral_finalize(const float* __restrict__ ws, float* __restrict__ out_real,
                float* __restrict__ outc) {
  size_t tid = (size_t)blockIdx.x * blockDim.x + threadIdx.x;   // < B*L*D
  int d = (int)(tid % D_DIM);
  size_t rest = tid / D_DIM;
  int l = (int)(rest % L_DIM);
  int i = l & (CH - 1);
  int c = l >> 4;
  int b = (int)(rest / L_DIM);
  float pr = ws[WS_PWRE + (i + 1) * D_DIM + d];
  float pi = ws[WS_PWIM + (i + 1) * D_DIM + d];
  size_t eidx = (size_t)(b * NC + c) * D_DIM + d;
  float er = ws[WS_E + 2 * eidx], ei = ws[WS_E + 2 * eidx + 1];
  float yr = outc[2 * tid], ym = outc[2 * tid + 1];
  float orr = yr + er * pr - ei * pi;
  float oim = ym + er * pi + ei * pr;
  out_real[tid] = orr;
  outc[2 * tid + 0] = orr;
  outc[2 * tid + 1] = oim;
}

// ---------------------------------------------------------------------------
extern "C" void kernel_launch(void* const* d_in, const int* in_sizes, int n_in,
                              void* d_out, int out_size, void* d_ws, size_t ws_size,
                              hipStream_t stream) {
  const float* x   = (const float*)d_in[0];   // [B,L,D]
  const float* hre = (const float*)d_in[1];   // [B,D]
  const float* him = (const float*)d_in[2];   // [B,D]
  const float* phr = (const float*)d_in[3];   // [D]
  const float* phi = (const float*)d_in[4];   // [D]
  const float* pir = (const float*)d_in[5];   // [D]
  const float* pii = (const float*)d_in[6];   // [D]
  (void)in_sizes; (void)n_in; (void)out_size; (void)ws_size;

  float* ws       = (float*)d_ws;
  float* out_real = (float*)d_out;                                  // [B,L,D] f32
  float* outc     = out_real + (size_t)B_DIM * L_DIM * D_DIM;       // [B,L,D] c64 interleaved

  spiral_setup<<<(D_DIM + 255) / 256, 256, 0, stream>>>(phr, phi, pir, pii, ws);

  spiral_chunk_wmma<<<B_DIM * (L_DIM / LT) * (D_DIM / DT), 256, 0, stream>>>(x, ws, outc);

  spiral_carry<<<(B_DIM * D_DIM + 255) / 256, 256, 0, stream>>>(
      hre, him, ws, outc, ws + WS_E);

  spiral_finalize<<<(int)(((size_t)B_DIM * L_DIM * D_DIM) / 256), 256, 0, stream>>>(
      ws, out_real, outc);
}